// GATNet_45303315038445
// MI455X (gfx1250) — compile-verified
//
#include <hip/hip_runtime.h>
#include <math.h>

typedef float v2f __attribute__((ext_vector_type(2)));
typedef float v8f __attribute__((ext_vector_type(8)));

#define TPB 256
#define NEG_SLOPE 0.2f

// ---------------------------------------------------------------------------
// Fill kernel (grid-stride)
// ---------------------------------------------------------------------------
__global__ void fill_kernel(float* __restrict__ p, float v, int n) {
    int i = blockIdx.x * blockDim.x + threadIdx.x;
    int stride = gridDim.x * blockDim.x;
    for (; i < n; i += stride) p[i] = v;
}

// ---------------------------------------------------------------------------
// GEMM via V_WMMA_F32_16X16X4_F32.  C[M x N] = A[M x K] * B[K x N]
// One wave per 16x16 output tile; K stepped by 4.
// f32 A-matrix 16x4 layout: lane m = lane&15, half-wave kh = lane>>4 picks
// K pair {2*kh, 2*kh+1}.  B 4x16: same K pair, column n = lane&15.
// C/D 16x16: VGPR r -> row r + 8*kh, col = lane&15.
// Out-of-range rows/cols are CLAMPED (not predicated) so the K loop runs
// with full EXEC and unconditional loads; stores are guarded.
// ---------------------------------------------------------------------------
__global__ void gemm_wmma_f32(const float* __restrict__ A,
                              const float* __restrict__ B,
                              float* __restrict__ C,
                              int M, int N, int K, int ldc) {
    int lane = threadIdx.x & 31;
    int wid  = (blockIdx.x * blockDim.x + threadIdx.x) >> 5;
    int ntn  = (N + 15) >> 4;
    int ntm  = (M + 15) >> 4;
    if (wid >= ntm * ntn) return;           // whole-wave uniform exit
    int tm = wid / ntn;
    int tn = wid - tm * ntn;

    int m  = lane & 15;
    int kh = lane >> 4;                     // 0 or 1: selects K pair
    int row = tm * 16 + m;                  // A row for this lane
    int rowc = (row < M) ? row : (M - 1);   // clamped (keeps EXEC full)
    int col = tn * 16 + m;                  // B/C column for this lane
    bool colOk = (col < N);
    int colc = colOk ? col : (N - 1);       // clamped

    // 8-byte aligned: K is even (multiple of 4) and 2*kh is even
    const float* ap = A + (size_t)rowc * K + 2 * kh;
    const float* bp = B + (size_t)(2 * kh) * N + colc;
    const size_t bstep = (size_t)4 * N;

    v8f acc = {};
#pragma unroll 4
    for (int kk = 0; kk < K; kk += 4) {
        v2f a = *(const v2f*)ap;            // one b64 load
        v2f b;
        b.x = bp[0];
        b.y = bp[N];
        acc = __builtin_amdgcn_wmma_f32_16x16x4_f32(
            /*neg_a=*/false, a, /*neg_b=*/false, b,
            /*c_mod=*/(short)0, acc, /*reuse_a=*/false, /*reuse_b=*/false);
        ap += 4;
        bp += bstep;
    }

#pragma unroll
    for (int r = 0; r < 8; ++r) {
        int orow = tm * 16 + r + 8 * kh;
        if (orow < M && colOk) C[(size_t)orow * ldc + col] = acc[r];
    }
}

// ---------------------------------------------------------------------------
// alpha_src[n] = dot(h[n], a_src); alpha_dst[n] = dot(h[n], a_dst)
// ---------------------------------------------------------------------------
__global__ void alpha_kernel(const float* __restrict__ h,
                             const float* __restrict__ avs,
                             const float* __restrict__ avd,
                             float* __restrict__ as_, float* __restrict__ ad_,
                             int Nn, int F) {
    int n = blockIdx.x * blockDim.x + threadIdx.x;
    if (n >= Nn) return;
    float s = 0.f, d = 0.f;
    const float* hr = h + (size_t)n * F;
#pragma unroll 4
    for (int f = 0; f < F; ++f) {
        float hv = hr[f];
        s += hv * avs[f];
        d += hv * avd[f];
    }
    as_[n] = s;
    ad_[n] = d;
}

// ---------------------------------------------------------------------------
// float atomic max via monotone int mapping
// ---------------------------------------------------------------------------
__device__ __forceinline__ void atomicMaxF32(float* addr, float val) {
    if (val >= 0.f)
        atomicMax((int*)addr, __float_as_int(val));
    else
        atomicMin((unsigned int*)addr, (unsigned int)__float_as_int(val));
}

// edge e in [0, E+Nn): e < E -> real edge, else self-loop (node e-E)
__global__ void edge_logit_max(const int* __restrict__ ei, int E, int Nn,
                               const float* __restrict__ as_,
                               const float* __restrict__ ad_,
                               float* __restrict__ ev, float* __restrict__ mx) {
    int e = blockIdx.x * blockDim.x + threadIdx.x;
    int ET = E + Nn;
    if (e >= ET) return;
    int s = (e < E) ? ei[e] : (e - E);
    int d = (e < E) ? ei[E + e] : (e - E);
    float v = as_[s] + ad_[d];
    v = (v > 0.f) ? v : NEG_SLOPE * v;      // leaky_relu
    ev[e] = v;
    atomicMaxF32(mx + d, v);
}

__global__ void edge_expsum(const int* __restrict__ ei, int E, int Nn,
                            const float* __restrict__ ev,
                            const float* __restrict__ mx,
                            float* __restrict__ ew, float* __restrict__ den) {
    int e = blockIdx.x * blockDim.x + threadIdx.x;
    int ET = E + Nn;
    if (e >= ET) return;
    int d = (e < E) ? ei[E + e] : (e - E);
    float w = expf(ev[e] - mx[d]);
    ew[e] = w;
    atomicAdd(den + d, w);
}

// one wave per edge: agg[dst] += h[src] * att  (features strided across lanes)
__global__ void edge_aggregate(const int* __restrict__ ei, int E, int Nn,
                               const float* __restrict__ h,
                               const float* __restrict__ ew,
                               const float* __restrict__ den,
                               float* __restrict__ agg, int F) {
    int lane = threadIdx.x & 31;
    int e = (blockIdx.x * blockDim.x + threadIdx.x) >> 5;
    int ET = E + Nn;
    if (e >= ET) return;
    int s = (e < E) ? ei[e] : (e - E);
    int d = (e < E) ? ei[E + e] : (e - E);
    float att = ew[e] / den[d];
    const float* hs = h + (size_t)s * F;
    float* ag = agg + (size_t)d * F;
    for (int f = lane; f < F; f += 32)
        atomicAdd(ag + f, hs[f] * att);
}

__global__ void bias_relu(const float* __restrict__ agg,
                          const float* __restrict__ b,
                          float* __restrict__ out, int Nn, int F) {
    int i = blockIdx.x * blockDim.x + threadIdx.x;
    if (i >= Nn * F) return;
    float v = agg[i] + b[i % F];
    out[i] = (v > 0.f) ? v : 0.f;
}

__global__ void bias_sigmoid(float* __restrict__ out,
                             const float* __restrict__ b, int Nn, int F) {
    int i = blockIdx.x * blockDim.x + threadIdx.x;
    if (i >= Nn * F) return;
    float v = out[i] + b[i % F];
    out[i] = 1.f / (1.f + expf(-v));
}

// ---------------------------------------------------------------------------
extern "C" void kernel_launch(void* const* d_in, const int* in_sizes, int n_in,
                              void* d_out, int out_size, void* d_ws, size_t ws_size,
                              hipStream_t stream) {
    const float* x     = (const float*)d_in[0];
    const int*   ei    = (const int*)d_in[1];
    const float* W1    = (const float*)d_in[2];
    const float* av_s1 = (const float*)d_in[3];
    const float* av_d1 = (const float*)d_in[4];
    const float* b1    = (const float*)d_in[5];
    const float* W2    = (const float*)d_in[6];
    const float* av_s2 = (const float*)d_in[7];
    const float* av_d2 = (const float*)d_in[8];
    const float* b2    = (const float*)d_in[9];

    const int HID = in_sizes[3];            // 64
    const int C   = in_sizes[7];            // 121
    const int FIN = in_sizes[2] / HID;      // 128
    const int Nn  = in_sizes[0] / FIN;      // 50000
    const int E   = in_sizes[1] / 2;        // 800000
    const int ET  = E + Nn;                 // edges + self loops

    float* w = (float*)d_ws;
    float* h1   = w; w += (size_t)Nn * HID;   // layer-1 features (reused post-relu)
    float* agg1 = w; w += (size_t)Nn * HID;
    float* h2   = w; w += (size_t)Nn * C;
    float* as_  = w; w += Nn;
    float* ad_  = w; w += Nn;
    float* mx   = w; w += Nn;
    float* den  = w; w += Nn;
    float* ev   = w; w += ET;
    float* ew   = w; w += ET;

    float* out = (float*)d_out;

    auto fillBlocks = [](int n) { int b = (n + TPB - 1) / TPB; return b > 4096 ? 4096 : b; };
    int edgeBlocks = (ET + TPB - 1) / TPB;
    int aggBlocks  = (ET * 32 + TPB - 1) / TPB;   // one wave per edge

    // ---------------- Layer 1 ----------------
    {
        int tiles = ((Nn + 15) / 16) * ((HID + 15) / 16);
        int gb = (tiles * 32 + TPB - 1) / TPB;
        gemm_wmma_f32<<<gb, TPB, 0, stream>>>(x, W1, h1, Nn, HID, FIN, HID);
    }
    alpha_kernel<<<(Nn + TPB - 1) / TPB, TPB, 0, stream>>>(h1, av_s1, av_d1, as_, ad_, Nn, HID);
    fill_kernel<<<fillBlocks(Nn), TPB, 0, stream>>>(mx, -INFINITY, Nn);
    fill_kernel<<<fillBlocks(Nn), TPB, 0, stream>>>(den, 0.f, Nn);
    fill_kernel<<<fillBlocks(Nn * HID), TPB, 0, stream>>>(agg1, 0.f, Nn * HID);
    edge_logit_max<<<edgeBlocks, TPB, 0, stream>>>(ei, E, Nn, as_, ad_, ev, mx);
    edge_expsum<<<edgeBlocks, TPB, 0, stream>>>(ei, E, Nn, ev, mx, ew, den);
    edge_aggregate<<<aggBlocks, TPB, 0, stream>>>(ei, E, Nn, h1, ew, den, agg1, HID);
    bias_relu<<<(Nn * HID + TPB - 1) / TPB, TPB, 0, stream>>>(agg1, b1, h1, Nn, HID);

    // ---------------- Layer 2 ----------------
    {
        int tiles = ((Nn + 15) / 16) * ((C + 15) / 16);
        int gb = (tiles * 32 + TPB - 1) / TPB;
        gemm_wmma_f32<<<gb, TPB, 0, stream>>>(h1, W2, h2, Nn, C, HID, C);
    }
    alpha_kernel<<<(Nn + TPB - 1) / TPB, TPB, 0, stream>>>(h2, av_s2, av_d2, as_, ad_, Nn, C);
    fill_kernel<<<fillBlocks(Nn), TPB, 0, stream>>>(mx, -INFINITY, Nn);
    fill_kernel<<<fillBlocks(Nn), TPB, 0, stream>>>(den, 0.f, Nn);
    fill_kernel<<<fillBlocks(Nn * C), TPB, 0, stream>>>(out, 0.f, Nn * C);
    edge_logit_max<<<edgeBlocks, TPB, 0, stream>>>(ei, E, Nn, as_, ad_, ev, mx);
    edge_expsum<<<edgeBlocks, TPB, 0, stream>>>(ei, E, Nn, ev, mx, ew, den);
    edge_aggregate<<<aggBlocks, TPB, 0, stream>>>(ei, E, Nn, h2, ew, den, out, C);
    bias_sigmoid<<<(Nn * C + TPB - 1) / TPB, TPB, 0, stream>>>(out, b2, Nn, C);
}